// MathildeGCN_69226282877222
// MI455X (gfx1250) — compile-verified
//
#include <hip/hip_runtime.h>
#include <hip/hip_bf16.h>

typedef __attribute__((ext_vector_type(2))) float v2f;
typedef __attribute__((ext_vector_type(8))) float v8f;

#define GN_NODES  100000
#define GN_EDGES  1600000
#define GN_GRAPHS 1000
#define GH        128

// ---------------------------------------------------------------------------
// Degree / normalization
// ---------------------------------------------------------------------------
__global__ void deg_init_kernel(float* __restrict__ deg, int n) {
    int i = blockIdx.x * blockDim.x + threadIdx.x;
    if (i < n) deg[i] = 1.0f;  // self-loop contributes 1 to target degree
}

__global__ void deg_edges_kernel(const int* __restrict__ ei, float* __restrict__ deg, int E) {
    int e = blockIdx.x * blockDim.x + threadIdx.x;
    if (e < E) unsafeAtomicAdd(deg + ei[E + e], 1.0f);  // dst row
}

__global__ void deg_finalize_kernel(float* __restrict__ deg, int n) {
    int i = blockIdx.x * blockDim.x + threadIdx.x;
    if (i < n) deg[i] = rsqrtf(deg[i]);  // deg >= 1 always (self-loops)
}

// ---------------------------------------------------------------------------
// GEMM: T = A @ W  (A: nrows x 128, W: 128 x 128), fused self-loop epilogue:
//   AGG = T * dinv^2   (message of the self-loop edge)
// 8 waves/block, 16 rows per wave, v_wmma_f32_16x16x4_f32, W pre-swizzled
// into B-fragment layout in LDS.
// ---------------------------------------------------------------------------
__global__ __launch_bounds__(256) void gemm_kernel(
    const float* __restrict__ A, const float* __restrict__ W,
    float* __restrict__ T, float* __restrict__ AGG,
    const float* __restrict__ dinv, int nrows) {
    // B-fragment-swizzled W: index ((tile*32 + j)*2 + hi)*16 + n  holds
    // { W[4j+2hi][16*tile+n], W[4j+2hi+1][16*tile+n] }
    __shared__ v2f sB[8192];  // 64 KB

    const int tid = threadIdx.x;
    for (int idx = tid; idx < 8192; idx += 256) {
        int n  = idx & 15;
        int hi = (idx >> 4) & 1;
        int j  = (idx >> 5) & 31;
        int tt = idx >> 10;
        int k   = 4 * j + 2 * hi;
        int col = 16 * tt + n;
        v2f v;
        v.x = W[k * GH + col];
        v.y = W[(k + 1) * GH + col];
        sB[idx] = v;
    }
    __syncthreads();

    const int lane = tid & 31;
    const int wave = tid >> 5;
    const int m    = lane & 15;
    const int hi   = lane >> 4;
    const int rowBase = blockIdx.x * 128 + wave * 16;
    int row  = rowBase + m;
    int rowc = row < nrows ? row : (nrows - 1);  // clamp: keeps EXEC uniform for WMMA

    // A fragments for 16 rows x 128 K, 32 k-steps of 4
    v2f a[32];
    const float* arow = A + (size_t)rowc * GH + 2 * hi;
#pragma unroll
    for (int j = 0; j < 32; ++j)
        a[j] = *(const v2f*)(arow + 4 * j);

#pragma unroll
    for (int tt = 0; tt < 8; ++tt) {
        v8f acc = {0.f, 0.f, 0.f, 0.f, 0.f, 0.f, 0.f, 0.f};
#pragma unroll
        for (int j = 0; j < 32; ++j) {
            v2f b = sB[((tt * 32 + j) * 2 + hi) * 16 + m];
            acc = __builtin_amdgcn_wmma_f32_16x16x4_f32(
                false, a[j], false, b, (short)0, acc, false, false);
        }
        // D layout: lane n = lane%16, VGPR r -> out row = r + 8*(lane/16)
        int col = 16 * tt + m;
#pragma unroll
        for (int r = 0; r < 8; ++r) {
            int orow = rowBase + r + 8 * hi;
            if (orow < nrows) {
                float val = acc[r];
                size_t off = (size_t)orow * GH + col;
                T[off] = val;
                float dv = dinv[orow];
                AGG[off] = val * dv * dv;  // self-loop message
            }
        }
    }
}

// ---------------------------------------------------------------------------
// Edge scatter: AGG[dst] += T[src] * (dinv[src]*dinv[dst]). One wave per edge,
// float4 per lane, native f32 atomics into L2.
// ---------------------------------------------------------------------------
__global__ __launch_bounds__(256) void edge_kernel(
    const int* __restrict__ ei, const float* __restrict__ T,
    float* __restrict__ AGG, const float* __restrict__ dinv, int E) {
    int wave = (int)((blockIdx.x * blockDim.x + threadIdx.x) >> 5);
    int lane = threadIdx.x & 31;
    if (wave >= E) return;
    int src = ei[wave];
    int dst = ei[E + wave];
    float norm = dinv[src] * dinv[dst];
    const float4* tv = (const float4*)(T + (size_t)src * GH);
    float4 v = tv[lane];
    float* out = AGG + (size_t)dst * GH + 4 * lane;
    unsafeAtomicAdd(out + 0, v.x * norm);
    unsafeAtomicAdd(out + 1, v.y * norm);
    unsafeAtomicAdd(out + 2, v.z * norm);
    unsafeAtomicAdd(out + 3, v.w * norm);
}

// ---------------------------------------------------------------------------
// Post: y = LN(AGG + bias)*gamma + beta ; optional ReLU ; optional residual.
// One wave per node, 4 channels per lane, __shfl_xor wave32 reductions.
// ---------------------------------------------------------------------------
__global__ __launch_bounds__(256) void post_kernel(
    const float* __restrict__ AGG, const float* __restrict__ bias,
    const float* __restrict__ gamma, const float* __restrict__ beta,
    float* __restrict__ Hbuf, int n, int do_relu, int do_res) {
    int node = (int)((blockIdx.x * blockDim.x + threadIdx.x) >> 5);
    int lane = threadIdx.x & 31;
    if (node >= n) return;
    int base = node * GH + 4 * lane;
    int c = 4 * lane;
    float v0 = AGG[base + 0] + bias[c + 0];
    float v1 = AGG[base + 1] + bias[c + 1];
    float v2 = AGG[base + 2] + bias[c + 2];
    float v3 = AGG[base + 3] + bias[c + 3];

    float s = v0 + v1 + v2 + v3;
    for (int off = 16; off >= 1; off >>= 1) s += __shfl_xor(s, off, 32);
    float mu = s * (1.0f / 128.0f);

    float d0 = v0 - mu, d1 = v1 - mu, d2 = v2 - mu, d3 = v3 - mu;
    float q = d0 * d0 + d1 * d1 + d2 * d2 + d3 * d3;
    for (int off = 16; off >= 1; off >>= 1) q += __shfl_xor(q, off, 32);
    float rs = rsqrtf(q * (1.0f / 128.0f) + 1e-5f);

    float y0 = d0 * rs * gamma[c + 0] + beta[c + 0];
    float y1 = d1 * rs * gamma[c + 1] + beta[c + 1];
    float y2 = d2 * rs * gamma[c + 2] + beta[c + 2];
    float y3 = d3 * rs * gamma[c + 3] + beta[c + 3];
    if (do_relu) {
        y0 = fmaxf(y0, 0.f); y1 = fmaxf(y1, 0.f);
        y2 = fmaxf(y2, 0.f); y3 = fmaxf(y3, 0.f);
    }
    if (do_res) {
        y0 += Hbuf[base + 0]; y1 += Hbuf[base + 1];
        y2 += Hbuf[base + 2]; y3 += Hbuf[base + 3];
    }
    Hbuf[base + 0] = y0; Hbuf[base + 1] = y1;
    Hbuf[base + 2] = y2; Hbuf[base + 3] = y3;
}

// ---------------------------------------------------------------------------
// Pooling: out[g] = mean over nodes of (h . lin_w) + lin_b
// ---------------------------------------------------------------------------
__global__ void pool_zero_kernel(float* __restrict__ s, float* __restrict__ cnt, int G) {
    int i = blockIdx.x * blockDim.x + threadIdx.x;
    if (i < G) { s[i] = 0.0f; cnt[i] = 0.0f; }
}

__global__ __launch_bounds__(256) void pool_nodes_kernel(
    const float* __restrict__ H, const int* __restrict__ batch,
    const float* __restrict__ lw, float* __restrict__ s,
    float* __restrict__ cnt, int n) {
    int node = (int)((blockIdx.x * blockDim.x + threadIdx.x) >> 5);
    int lane = threadIdx.x & 31;
    if (node >= n) return;
    int base = node * GH + 4 * lane;
    int c = 4 * lane;
    float d = H[base + 0] * lw[c + 0] + H[base + 1] * lw[c + 1] +
              H[base + 2] * lw[c + 2] + H[base + 3] * lw[c + 3];
    for (int off = 16; off >= 1; off >>= 1) d += __shfl_xor(d, off, 32);
    if (lane == 0) {
        int g = batch[node];
        unsafeAtomicAdd(s + g, d);
        unsafeAtomicAdd(cnt + g, 1.0f);
    }
}

__global__ void pool_final_kernel(const float* __restrict__ s, const float* __restrict__ cnt,
                                  const float* __restrict__ lb, float* __restrict__ out, int G) {
    int g = blockIdx.x * blockDim.x + threadIdx.x;
    if (g < G) out[g] = s[g] / fmaxf(cnt[g], 1.0f) + lb[0];
}

// ---------------------------------------------------------------------------
extern "C" void kernel_launch(void* const* d_in, const int* in_sizes, int n_in,
                              void* d_out, int out_size, void* d_ws, size_t ws_size,
                              hipStream_t stream) {
    const float* x     = (const float*)d_in[0];
    const int*   ei    = (const int*)d_in[1];
    const int*   batch = (const int*)d_in[2];
    const float* w1    = (const float*)d_in[3];
    const float* wsW   = (const float*)d_in[4];
    const float* bs    = (const float*)d_in[5];
    const float* gm    = (const float*)d_in[6];
    const float* bt    = (const float*)d_in[7];
    const float* lw    = (const float*)d_in[8];
    const float* lb    = (const float*)d_in[9];
    float* out = (float*)d_out;

    const int N = GN_NODES, E = GN_EDGES, G = GN_GRAPHS;

    float* wsf  = (float*)d_ws;
    float* Hbuf = wsf;                       // N*128
    float* T    = Hbuf + (size_t)N * GH;     // N*128
    float* AGG  = T + (size_t)N * GH;        // N*128
    float* dinv = AGG + (size_t)N * GH;      // N
    float* sg   = dinv + N;                  // G
    float* cg   = sg + G;                    // G

    // degree / rsqrt normalization (float atomics; exact < 2^24)
    deg_init_kernel<<<(N + 255) / 256, 256, 0, stream>>>(dinv, N);
    deg_edges_kernel<<<(E + 255) / 256, 256, 0, stream>>>(ei, dinv, E);
    deg_finalize_kernel<<<(N + 255) / 256, 256, 0, stream>>>(dinv, N);

    const int gemmBlocks = (N + 127) / 128;
    const int edgeBlocks = (E + 7) / 8;   // 8 waves/block, 1 edge/wave
    const int nodeBlocks = (N + 7) / 8;   // 8 waves/block, 1 node/wave

    for (int layer = 0; layer < 7; ++layer) {
        const float* inH = (layer == 0) ? x : Hbuf;
        const float* W   = (layer == 0) ? w1 : wsW + (size_t)(layer - 1) * GH * GH;
        gemm_kernel<<<gemmBlocks, 256, 0, stream>>>(inH, W, T, AGG, dinv, N);
        edge_kernel<<<edgeBlocks, 256, 0, stream>>>(ei, T, AGG, dinv, E);
        post_kernel<<<nodeBlocks, 256, 0, stream>>>(
            AGG, bs + layer * GH, gm + layer * GH, bt + layer * GH, Hbuf, N,
            (layer < 6) ? 1 : 0, (layer > 0) ? 1 : 0);
    }

    pool_zero_kernel<<<(G + 255) / 256, 256, 0, stream>>>(sg, cg, G);
    pool_nodes_kernel<<<nodeBlocks, 256, 0, stream>>>(Hbuf, batch, lw, sg, cg, N);
    pool_final_kernel<<<(G + 255) / 256, 256, 0, stream>>>(sg, cg, lb, out, G);
}